// StrictQuadraticFieldEnergyReadout_13872744366320
// MI455X (gfx1250) — compile-verified
//
#include <hip/hip_runtime.h>
#include <hip/hip_bf16.h>
#include <cstddef>

// ---------------------------------------------------------------------------
// StrictQuadraticFieldEnergyReadout, collapsed form:
//   energy[n] = s0[n] * dot(nf[n], Cp) + s1[n] * dot(nf[n], Dp)
//   s0 = (ci0-c0_0) * 0.01*ff0 ;  s1 = sum_m (ci_m-c0_m) * 0.01*ff_m  (m=1..3)
//   Cp = aD*aQ *  Wdown^T (W00 : wqa0 x wva0)          (128-vec)
//   Dp = aD*aQ/sqrt(3) * Wdown^T (W11 : wqa1 x wva1)   (128-vec)
// Memory-bound: ~560 B/node.  WMMA f32 16x16x4 folds the dual 128-dot per
// 16-node tile, staged through LDS (padded stride 132 -> conflict-free).
// ---------------------------------------------------------------------------

#define MIDN 32
#define NFN  128

#define ALPHA_QV_C  0.022097086912079608f   /* 1/sqrt(2*32*32) */
#define ALPHA_DOWN_C 0.015625f              /* 1/sqrt(32*128) = 1/64 */
#define INV_SQRT3_C 0.5773502691896258f

typedef __attribute__((ext_vector_type(2))) float v2f;
typedef __attribute__((ext_vector_type(8))) float v8f;

// ---------------- tiny weight-collapse kernel (one block) ------------------
__global__ __launch_bounds__(128)
void precompute_cd(const float* __restrict__ W00, const float* __restrict__ W11,
                   const float* __restrict__ Wd,
                   const float* __restrict__ wqa0, const float* __restrict__ wqa1,
                   const float* __restrict__ wva0, const float* __restrict__ wva1,
                   float* __restrict__ CD /* [256]: Cp | Dp */) {
    __shared__ float Ac[MIDN];
    __shared__ float Bc[MIDN];
    const int t = threadIdx.x;
    if (t < 64) {
        const int w = t & 31;
        const float* W  = (t < 32) ? W00  : W11;
        const float* wq = (t < 32) ? wqa0 : wqa1;
        const float* wv = (t < 32) ? wva0 : wva1;
        float s = 0.0f;
        for (int u = 0; u < MIDN; ++u) {
            const float qu = wq[u];
            for (int v = 0; v < MIDN; ++v)
                s += W[u * 1024 + v * 32 + w] * qu * wv[v];
        }
        if (t < 32) Ac[w] = s; else Bc[w] = s;
    }
    __syncthreads();
    // t = 0..127 -> output channel v
    float c = 0.0f, d = 0.0f;
    for (int u = 0; u < MIDN; ++u) {
        const float wdu = Wd[u * NFN + t];
        c += wdu * Ac[u];
        d += wdu * Bc[u];
    }
    const float sc = ALPHA_DOWN_C * ALPHA_QV_C;
    CD[t]       = sc * c;
    CD[NFN + t] = sc * INV_SQRT3_C * d;
}

// ---------------- main streaming kernel (WMMA per 16-node tile) ------------
#define WAVES 4
#define ROWS  16                       // nodes per wave tile
#define RSTR  (NFN + 4)                // 132 floats: conflict-free b64 reads
#define NODES_PER_BLOCK (WAVES * ROWS) // 64

__global__ __launch_bounds__(128)
void energy_kernel(const float* __restrict__ nf,
                   const float* __restrict__ c0,
                   const float* __restrict__ ci,
                   const float* __restrict__ ff,
                   const float* __restrict__ CD,
                   float* __restrict__ out, int nNodes) {
    __shared__ float tile[WAVES][ROWS][RSTR];   // 33,792 B
    __shared__ float wB[3][NFN + 2];            // Cp | Dp | zeros
    __shared__ float scr[WAVES][ROWS][2];       // per-node dotC/dotD

    const int t    = threadIdx.x;
    const int lane = t & 31;
    const int wave = t >> 5;
    const int n16  = lane & 15;
    const int hi   = lane >> 4;

    // B-fragment source table (broadcast zero row for lanes N>=2)
    if (t < NFN) {
        wB[0][t] = CD[t];
        wB[1][t] = CD[NFN + t];
        wB[2][t] = 0.0f;
    }

    const int blockBase = blockIdx.x * NODES_PER_BLOCK;
    const int waveBase  = blockBase + wave * ROWS;

    // Stage 16x128 f32 tile per wave: coalesced 512B per instruction.
    #pragma unroll 4
    for (int r = 0; r < ROWS; ++r) {
        int node = waveBase + r;
        if (node >= nNodes) node = nNodes - 1;          // clamp: keep EXEC full
        const float4 v = *(const float4*)(nf + (size_t)node * NFN + lane * 4);
        *(float4*)(&tile[wave][r][lane * 4]) = v;
    }
    __syncthreads();

    // D(16x16) = A(16 nodes x 128) * B(128 x 16), only N=0 (Cp) / N=1 (Dp) used.
    // A frag (16x4 f32): lane<16 -> K={0,1}, lane>=16 -> K={2,3}, M = lane&15.
    // B frag (4x16 f32): mirrored: N = lane&15, K split by lane half / vgpr.
    v8f acc = {};
    const int bRow = (n16 < 2) ? n16 : 2;
    #pragma unroll
    for (int kk = 0; kk < NFN / 4; ++kk) {
        const v2f a = *(const v2f*)(&tile[wave][n16][4 * kk + 2 * hi]);
        const v2f b = *(const v2f*)(&wB[bRow][4 * kk + 2 * hi]);
        acc = __builtin_amdgcn_wmma_f32_16x16x4_f32(
            /*neg_a=*/false, a, /*neg_b=*/false, b,
            /*c_mod=*/(short)0, acc, /*reuse_a=*/false, /*reuse_b=*/false);
    }

    // C/D layout: VGPR j, lanes 0-15 -> (M=j, N=lane); lanes 16-31 -> (M=j+8).
    if (n16 < 2) {
        #pragma unroll
        for (int j = 0; j < 8; ++j)
            scr[wave][hi * 8 + j][n16] = acc[j];
    }
    __syncthreads();

    // Finalize: one thread per node of this block.
    if (t < NODES_PER_BLOCK) {
        const int node = blockBase + t;
        if (node < nNodes) {
            const int w = t >> 4, l = t & 15;
            const float dotC = scr[w][l][0];
            const float dotD = scr[w][l][1];
            const float4 a0 = *(const float4*)(c0 + (size_t)node * 4);
            const float4 a1 = *(const float4*)(ci + (size_t)node * 4);
            const float4 fv = *(const float4*)(ff + (size_t)node * 4);
            const float s0 = (a1.x - a0.x) * (0.01f * fv.x);
            const float s1 = (a1.y - a0.y) * (0.01f * fv.y)
                           + (a1.z - a0.z) * (0.01f * fv.z)
                           + (a1.w - a0.w) * (0.01f * fv.w);
            out[node] = s0 * dotC + s1 * dotD;
        }
    }
}

extern "C" void kernel_launch(void* const* d_in, const int* in_sizes, int n_in,
                              void* d_out, int out_size, void* d_ws, size_t ws_size,
                              hipStream_t stream) {
    const float* nf   = (const float*)d_in[0];   // node_feats  [N,128]
    const float* c0   = (const float*)d_in[1];   // charges_0   [N,4]
    const float* ci   = (const float*)d_in[2];   // charges_induced [N,4]
    // d_in[3] edge_feats, d_in[4] edge_attrs: unused by reference
    const float* ff   = (const float*)d_in[5];   // field_feats [N,4]
    // d_in[6] edge_index, d_in[7] batch: unused by reference
    const float* wqa0 = (const float*)d_in[8];
    const float* wqa1 = (const float*)d_in[9];
    const float* wva0 = (const float*)d_in[10];
    const float* wva1 = (const float*)d_in[11];
    const float* W00  = (const float*)d_in[12];
    const float* W11  = (const float*)d_in[13];
    const float* Wd   = (const float*)d_in[14];
    float* out = (float*)d_out;
    float* CD  = (float*)d_ws;                    // 256 floats of scratch

    const int nNodes = in_sizes[0] / NFN;

    precompute_cd<<<1, 128, 0, stream>>>(W00, W11, Wd, wqa0, wqa1, wva0, wva1, CD);

    const int blocks = (nNodes + NODES_PER_BLOCK - 1) / NODES_PER_BLOCK;
    energy_kernel<<<blocks, 128, 0, stream>>>(nf, c0, ci, ff, CD, out, nNodes);
}